// MultiBackScatter_36094905155937
// MI455X (gfx1250) — compile-verified
//
#include <hip/hip_runtime.h>

// MultiBackScatter for MI455X (gfx1250).
//
// Reference collapses (unique permutation indices, zeros propagate) to:
//   out = zeros([V0, F]); out[idx0[idx1[idx2[i]]]] = x[i]  for i in [0, N3)
//
// Memory-bound: ~410 MB of zero stores dominates (~17.6 us at 23.3 TB/s).
// Zero-fill uses the CDNA5 async path: GLOBAL_STORE_ASYNC_FROM_LDS_B128
// streams a zeroed LDS block to global memory (ASYNCcnt-tracked, drained
// with s_wait_asynccnt) -- no zero VGPR quads needed per store.
// Scatter rows move with non-temporal b128 load/store (write-once data,
// output >> 192 MB L2).

typedef float v4f __attribute__((ext_vector_type(4)));

#ifndef MBS_F
#define MBS_F 64            // feature dim (reference: F = 64)
#endif
#define F4 (MBS_F / 4)      // float4s per row = 16

// ---------------------------------------------------------------------------
// Kernel 1: zero-fill via async store-from-LDS.
// Each block zeroes a 512 B LDS block once; every wave instruction then
// streams 32 lanes x 16 B = 512 B of zeros from LDS to a contiguous global
// region. Each thread covers 4 x v4f at stride blockDim.x (per-wave
// contiguous 512 B lines). Fast path: one range check for all 4 stores.
// ---------------------------------------------------------------------------
__global__ __launch_bounds__(256) void mbs_zero_fill_async(
    v4f* __restrict__ out4, long long n4) {
    __shared__ __align__(16) v4f zbuf[32];   // 512 B of zeros
    if (threadIdx.x < 32) {
        zbuf[threadIdx.x] = (v4f){0.0f, 0.0f, 0.0f, 0.0f};
    }
    __syncthreads();

    // Low 32 bits of a generic pointer to LDS == wave-relative LDS byte
    // address (aperture mapping), which is what VSRC of the async store wants.
    const unsigned lds_addr =
        (unsigned)(unsigned long long)&zbuf[threadIdx.x & 31u];

    const long long base =
        (long long)blockIdx.x * (4LL * blockDim.x) + threadIdx.x;

    if (base + 3LL * blockDim.x < n4) {
        // Fast path: all 4 stores in range, no per-store guard.
#pragma unroll
        for (int k = 0; k < 4; ++k) {
            const unsigned long long g =
                (unsigned long long)(out4 + base + (long long)k * blockDim.x);
            asm volatile("global_store_async_from_lds_b128 %0, %1, off"
                         :: "v"(g), "v"(lds_addr)
                         : "memory");
        }
    } else {
#pragma unroll
        for (int k = 0; k < 4; ++k) {
            const long long i = base + (long long)k * blockDim.x;
            if (i < n4) {
                const unsigned long long g = (unsigned long long)(out4 + i);
                asm volatile("global_store_async_from_lds_b128 %0, %1, off"
                             :: "v"(g), "v"(lds_addr)
                             : "memory");
            }
        }
    }

    // Drain ASYNCcnt before the wave ends (LDS must stay live until the
    // async engine has read it).
#if __has_builtin(__builtin_amdgcn_s_wait_asynccnt)
    __builtin_amdgcn_s_wait_asynccnt(0);
#else
    asm volatile("s_wait_asynccnt 0x0" ::: "memory");
#endif
}

// ---------------------------------------------------------------------------
// Kernel 2: composed-index scatter. 16 lanes cooperate on one source row
// (F = 64 floats = 16 float4s). Index chain: idx2 -> idx1 -> idx0.
// Indices are unique (reference uses permutations), so plain stores suffice
// (no atomics). x is read once and out rows written once -> NT both ways.
// ---------------------------------------------------------------------------
__global__ __launch_bounds__(256) void mbs_scatter(
    const v4f* __restrict__ x4,
    const int* __restrict__ idx0,
    const int* __restrict__ idx1,
    const int* __restrict__ idx2,
    v4f* __restrict__ out4,
    int n3) {
    const int t   = blockIdx.x * blockDim.x + threadIdx.x;
    const int row = t >> 4;     // 16 threads per row
    const int c   = t & 15;
    if (row >= n3) return;

    // Dependent gather chain (16 lanes redundantly load the same values ->
    // coalesced + L0/L2 hits, negligible traffic).
    const int j2 = idx2[row];
    const int j1 = idx1[j2];
    const int j0 = idx0[j1];

    const v4f v = __builtin_nontemporal_load(x4 + (long long)row * F4 + c);
    __builtin_nontemporal_store(v, out4 + (long long)j0 * F4 + c);
}

// ---------------------------------------------------------------------------
// Launch: zero-fill then scatter, both on `stream` (stream order guarantees
// the memset is complete before the scatter overwrites its rows).
// Inputs (setup_inputs order): x[f32], idx0[i32], idx1[i32], idx2[i32],
// v0, v1, v2 (scalars, unused -- sizes derived from in_sizes/out_size).
// ---------------------------------------------------------------------------
extern "C" void kernel_launch(void* const* d_in, const int* in_sizes, int n_in,
                              void* d_out, int out_size, void* d_ws, size_t ws_size,
                              hipStream_t stream) {
    const v4f* x4   = (const v4f*)d_in[0];
    const int* idx0 = (const int*)d_in[1];
    const int* idx1 = (const int*)d_in[2];
    const int* idx2 = (const int*)d_in[3];
    v4f*       out4 = (v4f*)d_out;

    const int       n3 = in_sizes[3];              // 25000 rows (idx2 count)
    const long long n4 = (long long)out_size / 4;  // output float4 count

    // Zero-fill: 4 x v4f per thread, 256 threads per block.
    {
        const long long per_block = 4LL * 256LL;
        const int blocks = (int)((n4 + per_block - 1) / per_block);  // 25000
        mbs_zero_fill_async<<<blocks, 256, 0, stream>>>(out4, n4);
    }

    // Scatter: n3 * 16 threads.
    {
        const int threads = n3 * F4;
        const int blocks  = (threads + 255) / 256;                   // 1563
        mbs_scatter<<<blocks, 256, 0, stream>>>(x4, idx0, idx1, idx2, out4, n3);
    }
}